// Col2Octree_29265907155619
// MI455X (gfx1250) — compile-verified
//
#include <hip/hip_runtime.h>
#include <cstdint>
#include <cstddef>

// col2octree scatter-add, MI455X (gfx1250, wave32).
//   data_in: [C=64, K=27, N] f32,  octree: [N, K] i32 (-1 = empty)
//   out:     [C, N] f32
//
// Fast path (needs 64*N*4 bytes of workspace):
//   1) zero outT[N][64] in d_ws
//   2) scatter: per block stage 32 nodes x 64 ch (per k) into LDS via async
//      global->LDS (transposing in flight), then line-coalesced
//      global_atomic_add_f32 into node-major outT (lane = channel ->
//      32 consecutive floats per wave32 atomic)
//   3) dense LDS-tiled transpose outT[N][64] -> out[64][N]
// Fallback: zero d_out, same staging, atomics straight into [C,N].

#define CCH 64
#define KK  27

#ifndef __has_builtin
#define __has_builtin(x) 0
#endif

#if __has_builtin(__builtin_amdgcn_global_load_async_to_lds_b32)
#define HAVE_ASYNC_LDS 1
#else
#define HAVE_ASYNC_LDS 0
#endif

typedef __attribute__((address_space(1))) int* gas_i32p;
typedef __attribute__((address_space(3))) int* las_i32p;

static __device__ __forceinline__ void wait_async_zero() {
#if HAVE_ASYNC_LDS
#if __has_builtin(__builtin_amdgcn_s_wait_asynccnt)
    __builtin_amdgcn_s_wait_asynccnt(0);
#else
    asm volatile("s_wait_asynccnt 0" ::: "memory");
#endif
#endif
}

// ------------------------------------------------------------------ probe ---
// Never launched. First function in the file so the disasm snippet shows the
// exact lowering of (a) global_load_async_to_lds_b32 and (b) unsafeAtomicAdd
// (want: global_atomic_add_f32, NOT a CAS loop).
__global__ __launch_bounds__(32) void probe_lowering(float* __restrict__ out,
                                                     const float* __restrict__ g) {
    __shared__ float l[32];
    float* lp = &l[threadIdx.x];
    const float* gp = &g[threadIdx.x];
#if HAVE_ASYNC_LDS
    __builtin_amdgcn_global_load_async_to_lds_b32((gas_i32p)(gp), (las_i32p)(lp), 0, 0);
#else
    *lp = *gp;
#endif
    wait_async_zero();
    __syncthreads();
    unsafeAtomicAdd(&out[threadIdx.x], l[threadIdx.x]);
}

// ---------------------------------------------------------------- scatter ---
template <bool NODE_MAJOR>
__device__ __forceinline__ void scatter_body(const float* __restrict__ data_in,
                                             const int* __restrict__ octree,
                                             float* __restrict__ out, int N) {
    __shared__ float tile[32 * 65];     // [hl][c], pad 65 -> conflict-free
    __shared__ int   idxAll[32 * KK];   // [hl][k] neighbor indices

    const int t  = threadIdx.x;
    const int h0 = blockIdx.x * 32;
    const size_t KN = (size_t)KK * (size_t)N;

    // Preload the 32 neighbor-table rows once: one contiguous 3456B region.
    {
        const size_t base  = (size_t)h0 * KK;
        const size_t limit = (size_t)N * KK;
        for (int e = t; e < 32 * KK; e += 256)
            idxAll[e] = (base + e < limit) ? octree[base + e] : -1;
    }
    __syncthreads();

    for (int k = 0; k < KK; ++k) {
        // ---- stage data_in[c, k, h0:h0+32] -> tile[hl][c] (transposed) ----
        // Per wave: c is uniform, hl = lane -> each async op covers one
        // contiguous 128B line of global memory.
#pragma unroll
        for (int it = 0; it < 8; ++it) {
            int e  = it * 256 + t;
            int c  = e >> 5;   // 0..63
            int hl = e & 31;   // 0..31
            int h  = h0 + hl;
            float* lp = &tile[hl * 65 + c];
            if (h < N) {
                const float* gp = data_in + (size_t)c * KN + (size_t)k * N + (size_t)h;
#if HAVE_ASYNC_LDS
                __builtin_amdgcn_global_load_async_to_lds_b32(
                    (gas_i32p)(gp), (las_i32p)(lp), 0, 0);
#else
                *lp = *gp;
#endif
            } else {
                *lp = 0.f;
            }
        }
        wait_async_zero();
        __syncthreads();

        // ---- scatter: 32 contributions x 64 channels ----
        // lane = channel: NODE_MAJOR gives 32 consecutive floats per wave32
        // -> line-coalesced atomic traffic into L2-resident outT.
#pragma unroll
        for (int p = 0; p < 8; ++p) {
            int j = p * 4 + (t >> 6);  // contribution slot 0..31
            int c = t & 63;            // channel
            int idx = idxAll[j * KK + k];
            if (idx >= 0) {
                float v = tile[j * 65 + c];
                if (NODE_MAJOR) {
                    unsafeAtomicAdd(&out[(size_t)idx * CCH + c], v);
                } else {
                    unsafeAtomicAdd(&out[(size_t)c * N + idx], v);
                }
            }
        }
        __syncthreads();
    }
}

__global__ __launch_bounds__(256) void scatter_node_major(const float* __restrict__ data_in,
                                                          const int* __restrict__ octree,
                                                          float* __restrict__ outT, int N) {
    scatter_body<true>(data_in, octree, outT, N);
}

__global__ __launch_bounds__(256) void scatter_chan_major(const float* __restrict__ data_in,
                                                          const int* __restrict__ octree,
                                                          float* __restrict__ out, int N) {
    scatter_body<false>(data_in, octree, out, N);
}

// -------------------------------------------------------------- transpose ---
// outT[N][64] -> out[64][N], 32-node x 64-channel LDS tiles, both sides coalesced.
__global__ __launch_bounds__(256) void transpose_kernel(const float* __restrict__ outT,
                                                        float* __restrict__ out, int N) {
    __shared__ float tile[CCH * 33];
    const int t  = threadIdx.x;
    const int n0 = blockIdx.x * 32;

#pragma unroll
    for (int it = 0; it < 8; ++it) {
        int e  = it * 256 + t;
        int nl = e >> 6;   // 0..31
        int c  = e & 63;   // 0..63
        int n  = n0 + nl;
        tile[c * 33 + nl] = (n < N) ? outT[(size_t)n * CCH + c] : 0.f;
    }
    __syncthreads();
#pragma unroll
    for (int it = 0; it < 8; ++it) {
        int e  = it * 256 + t;
        int c  = e >> 5;   // 0..63
        int nl = e & 31;   // 0..31
        int n  = n0 + nl;
        if (n < N) out[(size_t)c * N + n] = tile[c * 33 + nl];
    }
}

// ------------------------------------------------------------------- zero ---
__global__ __launch_bounds__(256) void zero_f4_kernel(float4* __restrict__ p, long n4) {
    long i = (long)blockIdx.x * blockDim.x + threadIdx.x;
    long stride = (long)gridDim.x * blockDim.x;
    float4 z; z.x = 0.f; z.y = 0.f; z.z = 0.f; z.w = 0.f;
    for (; i < n4; i += stride) p[i] = z;
}

// ----------------------------------------------------------------- launch ---
extern "C" void kernel_launch(void* const* d_in, const int* in_sizes, int n_in,
                              void* d_out, int out_size, void* d_ws, size_t ws_size,
                              hipStream_t stream) {
    const float* data_in = (const float*)d_in[0];
    const int*   octree  = (const int*)d_in[1];
    float*       out     = (float*)d_out;

    const int N = in_sizes[1] / KK;          // octree has N*K entries
    const int nblocks = (N + 31) / 32;

    const size_t need = (size_t)N * CCH * sizeof(float);
    if (ws_size >= need) {
        float* outT = (float*)d_ws;
        long n4 = ((long)N * CCH) / 4;
        int zb = (int)((n4 + 255) / 256); if (zb > 8192) zb = 8192;
        zero_f4_kernel<<<zb, 256, 0, stream>>>((float4*)outT, n4);
        scatter_node_major<<<nblocks, 256, 0, stream>>>(data_in, octree, outT, N);
        transpose_kernel<<<nblocks, 256, 0, stream>>>(outT, out, N);
    } else {
        long n4 = (long)out_size / 4;
        int zb = (int)((n4 + 255) / 256); if (zb > 8192) zb = 8192;
        zero_f4_kernel<<<zb, 256, 0, stream>>>((float4*)out, n4);
        scatter_chan_major<<<nblocks, 256, 0, stream>>>(data_in, octree, out, N);
    }
}